// Discriminator_61572651155699
// MI455X (gfx1250) — compile-verified
//
#include <hip/hip_runtime.h>
#include <math.h>

#define NN    50000
#define EE    800000
#define GG    16
#define NODE  128
#define HID   64
#define OUTF  128
#define HEADS 3
#define FTOT  192      /* HEADS*HID */
#define NEG_SLOPE 0.2f

typedef float v2f __attribute__((ext_vector_type(2)));
typedef float v8f __attribute__((ext_vector_type(8)));

__device__ __forceinline__ void atomicMaxF(float* addr, float val) {
    unsigned int* ua = (unsigned int*)addr;
    unsigned int cur = __float_as_uint(*addr);
    while (__uint_as_float(cur) < val) {
        unsigned int prev = atomicCAS(ua, cur, __float_as_uint(val));
        if (prev == cur) break;
        cur = prev;
    }
}

/* ---- init workspace: rst=0, mx=-inf, sum=0, pooled=0, cnt=0 ---- */
__global__ void k_init(float* __restrict__ rst, float* __restrict__ mx,
                       float* __restrict__ sm, float* __restrict__ pooled,
                       float* __restrict__ cnt) {
    int i = blockIdx.x * blockDim.x + threadIdx.x;
    if (i < NN * FTOT) rst[i] = 0.0f;
    if (i < NN * HEADS) { mx[i] = -__builtin_inff(); sm[i] = 0.0f; }
    if (i < GG * HID) pooled[i] = 0.0f;
    if (i < GG) cnt[i] = 0.0f;
}

/* ---- projection GEMM: feat[N,192] = h[N,128] @ fc_w^T, via f32 WMMA ----
 * One wave per 16-col strip: B strip (128x16 f32) lives in 64 VGPRs (v2f[32]);
 * A 16x128 tile staged in LDS (stride 132 to dodge bank conflicts).
 * WMMA layouts (ISA 7.12.2):
 *   A 16x4 f32:  lane m=lane&15, v2f = {K=klo, K=klo+1}, klo=(lane>>4)*2
 *   B 4x16 f32:  lane n=lane&15, v2f = {K=klo, K=klo+1}
 *   C/D 16x16:   VGPR r -> row r + (lane>=16 ? 8 : 0), col = lane&15
 */
__global__ void __launch_bounds__(32) k_proj(const float* __restrict__ h,
                                             const float* __restrict__ fcw,
                                             float* __restrict__ feat) {
    const int lane  = threadIdx.x;
    const int nbase = blockIdx.x * 16;
    const int m     = lane & 15;
    const int klo   = (lane >> 4) * 2;

    __shared__ __align__(16) float As[16 * 132];

    /* Preload B column strip: B[k][n] = fcw[n*NODE + k] */
    v2f Bk[32];
    {
        const float* fr = fcw + (size_t)(nbase + m) * NODE;
#pragma unroll
        for (int kk = 0; kk < 32; ++kk) {
            Bk[kk].x = fr[kk * 4 + klo];
            Bk[kk].y = fr[kk * 4 + klo + 1];
        }
    }

    const int rowTile0 = blockIdx.y * 8;
    for (int it = 0; it < 8; ++it) {
        int rowTile = rowTile0 + it;
        if (rowTile >= (NN / 16)) break;           /* uniform across wave */
        int rbase = rowTile * 16;

        /* cooperative A tile load -> LDS (float4, padded) */
        {
            const float4* h4  = (const float4*)(h + (size_t)rbase * NODE);
            float4*       As4 = (float4*)As;
#pragma unroll
            for (int r = 0; r < 16; ++r)
                As4[r * 33 + lane] = h4[r * 32 + lane];
        }
        __syncthreads();

        v8f acc = {0.f, 0.f, 0.f, 0.f, 0.f, 0.f, 0.f, 0.f};
#if __has_builtin(__builtin_amdgcn_wmma_f32_16x16x4_f32)
#pragma unroll
        for (int kk = 0; kk < 32; ++kk) {
            v2f a;
            a.x = As[m * 132 + kk * 4 + klo];
            a.y = As[m * 132 + kk * 4 + klo + 1];
            acc = __builtin_amdgcn_wmma_f32_16x16x4_f32(
                false, a, false, Bk[kk], (short)0, acc, false, false);
        }
#else
        /* scalar fallback producing identical D layout */
#pragma unroll
        for (int r = 0; r < 8; ++r) {
            int mm = r + ((lane >> 4) * 8);
            float s = 0.0f;
            for (int k = 0; k < NODE; ++k)
                s += As[mm * 132 + k] * fcw[(size_t)(nbase + m) * NODE + k];
            acc[r] = s;
        }
#endif
        __syncthreads();   /* protect As before next iteration's overwrite */

#pragma unroll
        for (int r = 0; r < 8; ++r) {
            int row = rbase + r + ((lane >> 4) * 8);
            feat[(size_t)row * FTOT + nbase + m] = acc[r];
        }
    }
}

/* ---- el/er attention coefficients per (node, head) ---- */
__global__ void k_attn(const float* __restrict__ feat, const float* __restrict__ al,
                       const float* __restrict__ ar, float* __restrict__ el,
                       float* __restrict__ er) {
    int i = blockIdx.x * blockDim.x + threadIdx.x;
    if (i >= NN * HEADS) return;
    int n = i / HEADS, hd = i % HEADS;
    const float* f = feat + (size_t)n * FTOT + hd * HID;
    float sl = 0.0f, sr = 0.0f;
#pragma unroll 8
    for (int d = 0; d < HID; ++d) {
        float v = f[d];
        sl += v * al[hd * HID + d];
        sr += v * ar[hd * HID + d];
    }
    el[i] = sl;
    er[i] = sr;
}

/* ---- edge logits + segment max over dst ---- */
__global__ void k_edge_logits(const int* __restrict__ src, const int* __restrict__ dst,
                              const float* __restrict__ el, const float* __restrict__ er,
                              float* __restrict__ ee, float* __restrict__ mx) {
    int e = blockIdx.x * blockDim.x + threadIdx.x;
    if (e >= EE) return;
    int s = src[e], d = dst[e];
#pragma unroll
    for (int hd = 0; hd < HEADS; ++hd) {
        float v = el[s * HEADS + hd] + er[d * HEADS + hd];
        v = (v > 0.0f) ? v : NEG_SLOPE * v;      /* leaky relu */
        ee[e * HEADS + hd] = v;
        atomicMaxF(&mx[d * HEADS + hd], v);
    }
}

/* ---- exp(e - max) + segment sum ---- */
__global__ void k_edge_exp(const int* __restrict__ dst, float* __restrict__ ee,
                           const float* __restrict__ mx, float* __restrict__ sm) {
    int e = blockIdx.x * blockDim.x + threadIdx.x;
    if (e >= EE) return;
    int d = dst[e];
#pragma unroll
    for (int hd = 0; hd < HEADS; ++hd) {
        float v = expf(ee[e * HEADS + hd] - mx[d * HEADS + hd]);
        ee[e * HEADS + hd] = v;
        atomicAdd(&sm[d * HEADS + hd], v);
    }
}

/* ---- normalize: alpha = exp / sum ---- */
__global__ void k_edge_norm(const int* __restrict__ dst, float* __restrict__ ee,
                            const float* __restrict__ sm) {
    int e = blockIdx.x * blockDim.x + threadIdx.x;
    if (e >= EE) return;
    int d = dst[e];
#pragma unroll
    for (int hd = 0; hd < HEADS; ++hd)
        ee[e * HEADS + hd] /= sm[d * HEADS + hd];
}

/* ---- scatter-weighted aggregation: rst[dst] += alpha * feat[src] ----
 * one thread per (edge, dim); 3 heads per thread. Memory-bound, L2-resident. */
__global__ void k_aggregate(const int* __restrict__ src, const int* __restrict__ dst,
                            const float* __restrict__ ee, const float* __restrict__ feat,
                            float* __restrict__ rst) {
    long long t = (long long)blockIdx.x * blockDim.x + threadIdx.x;
    if (t >= (long long)EE * HID) return;
    int e = (int)(t >> 6);
    int d = (int)(t & 63);
    int s = src[e], q = dst[e];
    const float* fs = feat + (size_t)s * FTOT + d;
    float*       rd = rst  + (size_t)q * FTOT + d;
    float a0 = ee[e * HEADS + 0], a1 = ee[e * HEADS + 1], a2 = ee[e * HEADS + 2];
    atomicAdd(rd,            a0 * fs[0]);
    atomicAdd(rd + HID,      a1 * fs[HID]);
    atomicAdd(rd + 2 * HID,  a2 * fs[2 * HID]);
}

/* ---- bias + head-mean + per-graph pooling accumulation ---- */
__global__ void k_node_pool(const float* __restrict__ rst, const float* __restrict__ bias,
                            const int* __restrict__ gid, float* __restrict__ pooled,
                            float* __restrict__ cnt) {
    int t = blockIdx.x * blockDim.x + threadIdx.x;
    if (t >= NN * HID) return;
    int n = t >> 6, d = t & 63;
    const float* r = rst + (size_t)n * FTOT;
    float v = (r[d] + r[HID + d] + r[2 * HID + d]
             + bias[d] + bias[HID + d] + bias[2 * HID + d]) * (1.0f / 3.0f);
    int g = gid[n];
    atomicAdd(&pooled[g * HID + d], v);
    if (d == 0) atomicAdd(&cnt[g], 1.0f);
}

/* ---- tiny MLP head: z1 = z@l1w^T + b; score = sigmoid([pooled,z1]@l2w^T + b) ---- */
__global__ void k_head(const float* __restrict__ pooled, const float* __restrict__ cnt,
                       const float* __restrict__ z, const float* __restrict__ l1w,
                       const float* __restrict__ l1b, const float* __restrict__ l2w,
                       const float* __restrict__ l2b, float* __restrict__ out) {
    int g = blockIdx.x;
    int j = threadIdx.x;  /* 0..127 */
    __shared__ float red[2 * HID];
    float c;
    if (j < HID) {
        float pv = pooled[g * HID + j] / fmaxf(cnt[g], 1.0f);
        c = pv * l2w[j];
    } else {
        int jj = j - HID;
        float s = l1b[jj];
        const float* zr = z + (size_t)g * OUTF;
        const float* wr = l1w + (size_t)jj * OUTF;
#pragma unroll 8
        for (int k = 0; k < OUTF; ++k) s += zr[k] * wr[k];
        c = s * l2w[HID + jj];
    }
    red[j] = c;
    __syncthreads();
    for (int st = HID; st > 0; st >>= 1) {
        if (j < st) red[j] += red[j + st];
        __syncthreads();
    }
    if (j == 0) out[g] = 1.0f / (1.0f + expf(-(red[0] + l2b[0])));
}

extern "C" void kernel_launch(void* const* d_in, const int* in_sizes, int n_in,
                              void* d_out, int out_size, void* d_ws, size_t ws_size,
                              hipStream_t stream) {
    (void)in_sizes; (void)n_in; (void)out_size; (void)ws_size;
    const float* h    = (const float*)d_in[0];
    const float* z    = (const float*)d_in[1];
    const int*   src  = (const int*)d_in[2];
    const int*   dst  = (const int*)d_in[3];
    const int*   gid  = (const int*)d_in[4];
    const float* fcw  = (const float*)d_in[5];
    const float* al   = (const float*)d_in[6];
    const float* ar   = (const float*)d_in[7];
    const float* bias = (const float*)d_in[8];
    const float* l1w  = (const float*)d_in[9];
    const float* l1b  = (const float*)d_in[10];
    const float* l2w  = (const float*)d_in[11];
    const float* l2b  = (const float*)d_in[12];
    float* out = (float*)d_out;

    float* ws = (float*)d_ws;
    size_t off = 0;
    float* feat   = ws + off; off += (size_t)NN * FTOT;
    float* rst    = ws + off; off += (size_t)NN * FTOT;
    float* el     = ws + off; off += (size_t)NN * HEADS;
    float* er     = ws + off; off += (size_t)NN * HEADS;
    float* mx     = ws + off; off += (size_t)NN * HEADS;
    float* sm     = ws + off; off += (size_t)NN * HEADS;
    float* ee     = ws + off; off += (size_t)EE * HEADS;
    float* pooled = ws + off; off += (size_t)GG * HID;
    float* cnt    = ws + off; off += (size_t)GG;

    k_init<<<(NN * FTOT + 255) / 256, 256, 0, stream>>>(rst, mx, sm, pooled, cnt);

    dim3 gproj(FTOT / 16, (NN / 16 + 7) / 8);
    k_proj<<<gproj, 32, 0, stream>>>(h, fcw, feat);

    k_attn<<<(NN * HEADS + 255) / 256, 256, 0, stream>>>(feat, al, ar, el, er);
    k_edge_logits<<<(EE + 255) / 256, 256, 0, stream>>>(src, dst, el, er, ee, mx);
    k_edge_exp<<<(EE + 255) / 256, 256, 0, stream>>>(dst, ee, mx, sm);
    k_edge_norm<<<(EE + 255) / 256, 256, 0, stream>>>(dst, ee, sm);

    long long aggT = (long long)EE * HID;
    k_aggregate<<<(unsigned)((aggT + 255) / 256), 256, 0, stream>>>(src, dst, ee, feat, rst);

    k_node_pool<<<(NN * HID + 255) / 256, 256, 0, stream>>>(rst, bias, gid, pooled, cnt);
    k_head<<<GG, 2 * HID, 0, stream>>>(pooled, cnt, z, l1w, l1b, l2w, l2b, out);
}